// BOW_model_9474697855288
// MI455X (gfx1250) — compile-verified
//
#include <hip/hip_runtime.h>
#include <hip/hip_bf16.h>

#define VOCAB        100000
#define HID          256
#define BATCH        4096
#define TOTAL_TOKENS 819200
#define BN_EPS       1e-5f

typedef __attribute__((ext_vector_type(2))) float v2f;
typedef __attribute__((ext_vector_type(8))) float v8f;

// -------------------------------------------------------------------------
// Kernel 1: ragged EmbeddingBag-sum.  Wave32: one wave per 128-token chunk,
// lane owns 8 consecutive features (2 x float4 -> full 1KB coalesced row).
// segment_ids are sorted, so we accumulate in registers and only flush via
// atomics when the segment id changes inside the chunk (plus once at end).
// -------------------------------------------------------------------------
__device__ __forceinline__ void flush_row(float* dst, const float4& a0, const float4& a1) {
    atomicAdd(dst + 0, a0.x); atomicAdd(dst + 1, a0.y);
    atomicAdd(dst + 2, a0.z); atomicAdd(dst + 3, a0.w);
    atomicAdd(dst + 4, a1.x); atomicAdd(dst + 5, a1.y);
    atomicAdd(dst + 6, a1.z); atomicAdd(dst + 7, a1.w);
}

__global__ void k_embbag(const int* __restrict__ tok,
                         const int* __restrict__ seg,
                         const float* __restrict__ emb,
                         float* __restrict__ seg_sum,
                         float* __restrict__ counts) {
    const int TPW  = 128;                                   // tokens per wave
    int wave = blockIdx.x * 8 + (threadIdx.x >> 5);         // 8 waves / block
    int lane = threadIdx.x & 31;
    long start = (long)wave * TPW;
    if (start >= TOTAL_TOKENS) return;
    long end = start + TPW; if (end > TOTAL_TOKENS) end = TOTAL_TOKENS;

    const int f0 = lane * 8;
    float4 a0 = make_float4(0.f, 0.f, 0.f, 0.f);
    float4 a1 = make_float4(0.f, 0.f, 0.f, 0.f);
    int curSeg = seg[start];
    int cnt = 0;

    for (long t = start; t < end; ++t) {
        int s = seg[t];
        if (s != curSeg) {
            if (cnt > 0) {
                flush_row(seg_sum + (long)curSeg * HID + f0, a0, a1);
                if (lane == 0) atomicAdd(counts + curSeg, (float)cnt);
            }
            a0 = make_float4(0.f, 0.f, 0.f, 0.f);
            a1 = make_float4(0.f, 0.f, 0.f, 0.f);
            cnt = 0;
            curSeg = s;
        }
        int tk = tok[t];
        const float4* row = (const float4*)(emb + (long)tk * HID + f0);
        float4 e0 = row[0];
        float4 e1 = row[1];
        if (t + 1 < end) {  // global_prefetch_b8 on next gathered row
            __builtin_prefetch(emb + (long)tok[t + 1] * HID + f0, 0, 1);
        }
        a0.x += e0.x; a0.y += e0.y; a0.z += e0.z; a0.w += e0.w;
        a1.x += e1.x; a1.y += e1.y; a1.z += e1.z; a1.w += e1.w;
        ++cnt;
    }
    if (cnt > 0) {
        flush_row(seg_sum + (long)curSeg * HID + f0, a0, a1);
        if (lane == 0) atomicAdd(counts + curSeg, (float)cnt);
    }
}

// -------------------------------------------------------------------------
// Kernel 2: h = (seg_sum / max(count,1)) @ W1^T + b1 using
// V_WMMA_F32_16X16X4_F32 (exact fp32).  Block = 16 M-rows; bow tile staged
// in LDS with a 260-float row stride (260 % 64 == 4 -> lanes stride 4 banks,
// all 64 banks hit once per ds_load_b64).  Each wave owns one 16-wide N-tile.
//
// Per ISA layouts (16x16x4 f32):
//  A (16x4): lane&15 = M, (lane>>4) selects K pair {0,1}/{2,3}, VGPR = K in pair
//  B (4x16): column-major -> lane&15 = N, same K mapping as A
//  C/D:      lane&15 = N, VGPR i -> M = (lane>>4)*8 + i
// -------------------------------------------------------------------------
__global__ void k_gemm_wmma(const float* __restrict__ seg_sum,
                            const float* __restrict__ counts,
                            const float* __restrict__ W1,
                            const float* __restrict__ b1,
                            float* __restrict__ h) {
    __shared__ float sA[16 * 260];
    const int m0 = blockIdx.x * 16;
    const int t  = threadIdx.x;

    // Stage bow tile (fused divide-by-count) into LDS: thread t loads 16 floats.
    {
        int row = t >> 4;
        int colBase = (t & 15) * 16;
        float c   = counts[m0 + row];
        float inv = 1.0f / fmaxf(c, 1.0f);
        const float* src = seg_sum + (long)(m0 + row) * HID + colBase;
        float* dst = sA + row * 260 + colBase;
        #pragma unroll
        for (int j = 0; j < 16; j += 4) {
            float4 v = *(const float4*)(src + j);
            dst[j + 0] = v.x * inv; dst[j + 1] = v.y * inv;
            dst[j + 2] = v.z * inv; dst[j + 3] = v.w * inv;
        }
    }
    __syncthreads();

    const int waveId = t >> 5;
    const int lane   = t & 31;
    const int nBase  = (blockIdx.y * 8 + waveId) * 16;
    const int mn     = lane & 15;          // M for A fetch, N for B fetch & D store
    const int kOff   = (lane >> 4) << 1;   // 0 or 2

    const float* aRow = sA + mn * 260 + kOff;                     // LDS, bank-spread
    const float* bRow = W1 + (long)(nBase + mn) * HID + kOff;     // B col-major = W1 row

    v8f acc = {};
    #pragma unroll 8
    for (int k0 = 0; k0 < HID; k0 += 4) {
        v2f a, b;
        a.x = aRow[k0]; a.y = aRow[k0 + 1];
        b.x = bRow[k0]; b.y = bRow[k0 + 1];
        acc = __builtin_amdgcn_wmma_f32_16x16x4_f32(
            /*neg_a=*/false, a, /*neg_b=*/false, b,
            /*c_mod=*/(short)0, acc, /*reuse_a=*/false, /*reuse_b=*/false);
    }

    const int mHalf = (lane >> 4) * 8;
    const float bias = b1[nBase + mn];
    #pragma unroll
    for (int i = 0; i < 8; ++i) {
        h[(long)(m0 + mHalf + i) * HID + nBase + mn] = acc[i] + bias;
    }
}

// -------------------------------------------------------------------------
// Kernel 3: per-feature BatchNorm statistics over the 4096-row batch.
// One block per feature; mu and rstd = rsqrt(var + eps).
// -------------------------------------------------------------------------
__global__ void k_bnstats(const float* __restrict__ h,
                          float* __restrict__ mu,
                          float* __restrict__ rstd) {
    __shared__ float s1[256];
    __shared__ float s2[256];
    const int j = blockIdx.x;
    const int t = threadIdx.x;
    float sum = 0.f, sq = 0.f;
    for (int r = t; r < BATCH; r += 256) {
        float v = h[(long)r * HID + j];
        sum += v; sq += v * v;
    }
    s1[t] = sum; s2[t] = sq;
    __syncthreads();
    for (int off = 128; off > 0; off >>= 1) {
        if (t < off) { s1[t] += s1[t + off]; s2[t] += s2[t + off]; }
        __syncthreads();
    }
    if (t == 0) {
        float m   = s1[0] * (1.0f / BATCH);
        float var = s2[0] * (1.0f / BATCH) - m * m;   // biased var
        mu[j]   = m;
        rstd[j] = rsqrtf(var + BN_EPS);
    }
}

// -------------------------------------------------------------------------
// Kernel 4: normalize + gamma/beta + ReLU + dot(W2) + b2 -> logits, then
// mean BCE-with-logits accumulated atomically into out[0].
// One wave per batch row; lane owns 8 features; wave32 shfl_xor reduction.
// -------------------------------------------------------------------------
__global__ void k_head(const float* __restrict__ h,
                       const float* __restrict__ mu,
                       const float* __restrict__ rstd,
                       const float* __restrict__ gamma,
                       const float* __restrict__ beta,
                       const float* __restrict__ W2,
                       const float* __restrict__ b2,
                       const float* __restrict__ labels,
                       float* __restrict__ out) {   // out[0]=loss, out[1..BATCH]=logits
    int row  = blockIdx.x * 8 + (threadIdx.x >> 5);
    int lane = threadIdx.x & 31;
    if (row >= BATCH) return;
    const int f0 = lane * 8;
    float part = 0.f;
    #pragma unroll
    for (int j = 0; j < 8; ++j) {
        int f = f0 + j;
        float v  = h[(long)row * HID + f];
        float hn = (v - mu[f]) * rstd[f] * gamma[f] + beta[f];
        hn = fmaxf(hn, 0.f);
        part += hn * W2[f];
    }
    #pragma unroll
    for (int off = 16; off > 0; off >>= 1)
        part += __shfl_xor(part, off, 32);
    if (lane == 0) {
        float logit = part + b2[0];
        out[1 + row] = logit;
        float y = labels[row];
        float term = fmaxf(logit, 0.f) - logit * y + log1pf(expf(-fabsf(logit)));
        atomicAdd(out, term * (1.0f / BATCH));
    }
}

// -------------------------------------------------------------------------
extern "C" void kernel_launch(void* const* d_in, const int* in_sizes, int n_in,
                              void* d_out, int out_size, void* d_ws, size_t ws_size,
                              hipStream_t stream) {
    const int*   tok    = (const int*)d_in[0];
    const int*   seg    = (const int*)d_in[1];
    const float* labels = (const float*)d_in[2];
    const float* emb    = (const float*)d_in[3];
    const float* W1     = (const float*)d_in[4];
    const float* b1     = (const float*)d_in[5];
    const float* gamma  = (const float*)d_in[6];
    const float* beta   = (const float*)d_in[7];
    const float* W2     = (const float*)d_in[8];
    const float* b2     = (const float*)d_in[9];
    float* out = (float*)d_out;

    // Workspace layout (all fp32): seg_sum | counts | h | mu | rstd
    float* seg_sum = (float*)d_ws;
    float* counts  = seg_sum + (size_t)BATCH * HID;
    float* hbuf    = counts + BATCH;
    float* mu      = hbuf + (size_t)BATCH * HID;
    float* rstd    = mu + HID;

    // Zero the accumulation regions (graph-capturable async memsets).
    hipMemsetAsync(seg_sum, 0, ((size_t)BATCH * HID + BATCH) * sizeof(float), stream);
    hipMemsetAsync(out, 0, sizeof(float), stream);

    // 1) gather + segment sum:  819200 tokens / (128 per wave * 8 waves) = 800 blocks
    k_embbag<<<TOTAL_TOKENS / (128 * 8), 256, 0, stream>>>(tok, seg, emb, seg_sum, counts);
    // 2) WMMA GEMM:  256 M-tiles x (2 blocks * 8 waves = 16 N-tiles)
    k_gemm_wmma<<<dim3(BATCH / 16, 2), 256, 0, stream>>>(seg_sum, counts, W1, b1, hbuf);
    // 3) BN stats: one block per feature
    k_bnstats<<<HID, 256, 0, stream>>>(hbuf, mu, rstd);
    // 4) head: one wave per row
    k_head<<<BATCH / 8, 256, 0, stream>>>(hbuf, mu, rstd, gamma, beta, W2, b2, labels, out);
}